// Diffusion_57853209477679
// MI455X (gfx1250) — compile-verified
//
#include <hip/hip_runtime.h>

#define NN 2048            // n_nodes
#define CC 512             // channels
#define CC2 (2 * CC)       // pitch of one row-pair in the swizzled V layout
#define KTERMS 12          // ||tL|| ~ 0.2 -> term 12 ~ 1e-13

typedef __attribute__((ext_vector_type(2))) float v2f;
typedef __attribute__((ext_vector_type(8))) float v8f;

// Swizzled Taylor-vector layout: element (k, n) lives at
//   Vp[(k>>1)*CC2 + 2*n + (k&1)]
// so a B fragment pair (V[kk][n], V[kk+1][n]) (kk even) is ONE aligned b64 load.

// y = x ; vp = swizzle(x)
__global__ __launch_bounds__(256) void diff_init(const float* __restrict__ x,
                                                 float* __restrict__ y,
                                                 float* __restrict__ vp) {
  unsigned i = blockIdx.x * 256u + threadIdx.x;
  float d = x[i];
  y[i] = d;
  unsigned row = i / CC;
  unsigned col = i % CC;
  vp[(row >> 1) * CC2 + 2u * col + (row & 1u)] = d;
}

// One Taylor step:  W = (-t/k) * (L @ V);  Y += W
// Block: 256 threads = 8 wave32 arranged 2x4 -> 64x128 output tile.
// Each wave: 32x32 tile = 2x2 WMMA accumulators; per k-step (K=4):
//   2 A b64 loads + 2 B b64 loads -> 4 v_wmma_f32_16x16x4_f32  (1:1 vmem:wmma)
__global__ __launch_bounds__(256) void diff_gemm(const float* __restrict__ Lm,
                                                 const float* __restrict__ Vp,   // swizzled in
                                                 float* __restrict__ Wp,         // swizzled out
                                                 float* __restrict__ Yacc,       // row-major
                                                 const float* __restrict__ tptr,
                                                 int kterm) {
  const unsigned lane = threadIdx.x & 31u;
  const unsigned wave = threadIdx.x >> 5;
  const unsigned rowBase = blockIdx.x * 64u + (wave >> 2) * 32u;
  const unsigned colBase = blockIdx.y * 128u + (wave & 3u) * 32u;

  const unsigned lm = lane & 15u;   // 0..15
  const unsigned lh = lane >> 4;    // 0..1 (half-wave -> K pair select)

  v8f acc00 = {}, acc01 = {}, acc10 = {}, acc11 = {};

  // A (16x4 f32): lanes 0-15 -> row M=lane, K pair (k0, k0+1);
  //               lanes 16-31 -> row M=lane-16, K pair (k0+2, k0+3).
  const float* a0base = Lm + (size_t)(rowBase + lm) * NN + 2u * lh;
  const float* a1base = a0base + (size_t)16u * NN;
  // B (4x16 f32) from swizzled Vp: one b64 per fragment.
  const float* b0base = Vp + (size_t)lh * CC2 + 2u * (colBase + lm);
  const float* b1base = b0base + 32u;

  for (unsigned k0 = 0; k0 < NN; k0 += 4u) {
    const size_t aoff = k0;                    // 4 floats per k-step
    const size_t boff = (size_t)(k0 >> 1) * CC2;  // 2 row-pairs per k-step

    v2f a0 = *(const v2f*)(a0base + aoff);
    v2f a1 = *(const v2f*)(a1base + aoff);
    v2f b0 = *(const v2f*)(b0base + boff);
    v2f b1 = *(const v2f*)(b1base + boff);

    acc00 = __builtin_amdgcn_wmma_f32_16x16x4_f32(false, a0, false, b0,
                                                  (short)0, acc00, false, false);
    acc01 = __builtin_amdgcn_wmma_f32_16x16x4_f32(false, a0, false, b1,
                                                  (short)0, acc01, false, false);
    acc10 = __builtin_amdgcn_wmma_f32_16x16x4_f32(false, a1, false, b0,
                                                  (short)0, acc10, false, false);
    acc11 = __builtin_amdgcn_wmma_f32_16x16x4_f32(false, a1, false, b1,
                                                  (short)0, acc11, false, false);
  }

  float tt = tptr[0];
  tt = fmaxf(tt, 1e-8f);
  const float alpha = -tt / (float)kterm;

  // C/D layout: VGPR r -> rows (r, r+8) split across lane halves, col = lane%16.
  auto epilogue = [&](const v8f& acc, unsigned ti, unsigned tj) {
    const unsigned rowTop = rowBase + 16u * ti + 8u * lh;
    const unsigned col    = colBase + 16u * tj + lm;
#pragma unroll
    for (int r = 0; r < 8; ++r) {
      const unsigned row = rowTop + (unsigned)r;
      const float w = alpha * acc[r];
      Yacc[(size_t)row * CC + col] += w;   // element owned by exactly this thread
      Wp[(size_t)(row >> 1) * CC2 + 2u * col + (row & 1u)] = w;
    }
  };
  epilogue(acc00, 0u, 0u);
  epilogue(acc01, 0u, 1u);
  epilogue(acc10, 1u, 0u);
  epilogue(acc11, 1u, 1u);
}

extern "C" void kernel_launch(void* const* d_in, const int* in_sizes, int n_in,
                              void* d_out, int out_size, void* d_ws, size_t ws_size,
                              hipStream_t stream) {
  (void)in_sizes; (void)n_in; (void)out_size; (void)ws_size;

  const float* x  = (const float*)d_in[0];   // [N, C]
  const float* Lm = (const float*)d_in[1];   // [N, N]
  const float* t  = (const float*)d_in[2];   // [1]

  float* Y = (float*)d_out;                  // [N, C] row-major
  float* V = (float*)d_ws;                   // swizzled Taylor vector ping
  float* W = V + (size_t)NN * CC;            // swizzled Taylor vector pong

  // y = x ; v = swizzle(x)
  diff_init<<<dim3((NN * CC) / 256), dim3(256), 0, stream>>>(x, Y, V);

  // v <- (-t/k) L v ; y += v   for k = 1..KTERMS
  dim3 grid(NN / 64, CC / 128);              // 32 x 4 blocks, 8 waves each
  for (int k = 1; k <= KTERMS; ++k) {
    diff_gemm<<<grid, dim3(256), 0, stream>>>(Lm, V, W, Y, t, k);
    float* tmp = V; V = W; W = tmp;
  }
}